// FBControls_74680891343520
// MI455X (gfx1250) — compile-verified
//
#include <hip/hip_runtime.h>
#include <math.h>

typedef __attribute__((ext_vector_type(2))) float v2f;
typedef __attribute__((ext_vector_type(8))) float v8f;

#define B_   16
#define TT_  128
#define TS_  256
#define C_   512

// output layout (floats), concatenated in reference return order
#define COFF 0
#define GOFF (B_*(TT_-1)*TS_*2)            // controls: 1,040,384
#define ROFF (GOFF + B_*TT_*TS_)           // gamma:    +524,288
#define WOFF (ROFF + B_*(TT_-1)*TS_)       // read:     +520,192

// ---------------------------------------------------------------------------
// Kernel 1: gate GEMV via V_WMMA_F32_16X16X4_F32.
// A = gate_w broadcast over M (16x4), B = 16 obs rows (4x16). D[m,n]=y[n] ∀m.
// One wave per (b, t<127, s-tile of 16). 512/4 = 128 WMMAs per wave.
// ---------------------------------------------------------------------------
__global__ __launch_bounds__(256) void gate_kernel(
    const float* __restrict__ obs, const float* __restrict__ gw,
    const float* __restrict__ gb, float* __restrict__ out)
{
  const int wid   = (blockIdx.x * 256 + threadIdx.x) >> 5;   // wave id
  const int lane  = threadIdx.x & 31;
  const int stile = wid & 15;                                // 16 tiles of 16 s-rows
  const int bt    = wid >> 4;                                // 0 .. 16*127-1
  const int t     = bt % (TT_-1);
  const int b     = bt / (TT_-1);
  const int m     = lane & 15;
  const int h     = lane >> 4;

  const float* __restrict__ row =
      obs + ((size_t)((b*TT_ + t)*TS_ + stile*16 + m)) * C_;

  v8f acc = {0.f,0.f,0.f,0.f,0.f,0.f,0.f,0.f};
  #pragma unroll 4
  for (int cb = 0; cb < C_; cb += 4) {
    v2f bfrag = *(const v2f*)(row + cb + 2*h);   // obs tile (B operand)
    v2f afrag = *(const v2f*)(gw  + cb + 2*h);   // weights broadcast (A operand)
    acc = __builtin_amdgcn_wmma_f32_16x16x4_f32(
        false, afrag, false, bfrag, (short)0, acc, false, false);
  }

  float x = acc[0] + gb[0];                      // D identical across VGPRs/lanes-per-row
  float s = (x >= 0.f) ? (-log1pf(expf(-x))) : (x - log1pf(expf(x)));
  if (h == 0) {                                  // lanes 0-15: one float2 per s-row
    size_t o = ((size_t)((b*(TT_-1) + t)*TS_ + stile*16 + m)) * 2;
    float2 vv; vv.x = s; vv.y = s - x;
    *(float2*)(out + COFF + o) = vv;
  }
}

// ---------------------------------------------------------------------------
// Kernel 2: forward DP. One wave32 per batch; lane owns 8 contiguous s.
// fs[t+1][s] = Cf[s] + cummin_{j<=s}( fs[t][j] - scores[t,j] - Cf[j] )
// ---------------------------------------------------------------------------
__global__ __launch_bounds__(32) void fs_kernel(
    const float* __restrict__ scores, float* __restrict__ fs)
{
  const int b = blockIdx.x, lane = threadIdx.x;
  const int s0 = lane * 8;
  float Cf[8], f[8];
  #pragma unroll
  for (int j = 0; j < 8; ++j) {
    float sv = (float)(s0 + j);
    Cf[j] = ((sv+1.f)*(sv+2.f)*0.5f - 1.f) * (1.f/(float)TS_);
    f[j]  =  (sv+1.f)*(sv+2.f)*0.5f        * (1.f/(float)TS_);
  }
  float* fsb = fs + (size_t)b * (TT_+1) * TS_;
  const float* scb = scores + (size_t)b * TT_ * TS_;
  { float4 o0={f[0],f[1],f[2],f[3]}, o1={f[4],f[5],f[6],f[7]};
    *(float4*)(fsb + s0) = o0; *(float4*)(fsb + s0 + 4) = o1; }

  float4 sc0 = *(const float4*)(scb + s0);
  float4 sc1 = *(const float4*)(scb + s0 + 4);
  for (int t = 0; t < TT_; ++t) {
    float4 n0 = {}, n1 = {};
    if (t + 1 < TT_) {                       // prefetch next row
      n0 = *(const float4*)(scb + (t+1)*TS_ + s0);
      n1 = *(const float4*)(scb + (t+1)*TS_ + s0 + 4);
    }
    float scv[8] = {sc0.x,sc0.y,sc0.z,sc0.w,sc1.x,sc1.y,sc1.z,sc1.w};
    float g[8]; float run = __builtin_inff();
    #pragma unroll
    for (int j = 0; j < 8; ++j) {            // local inclusive cummin
      float gv = f[j] - scv[j] - Cf[j];      // sc_dp = -scores
      run = fminf(run, gv); g[j] = run;
    }
    float v = run;                           // wave inclusive min-scan
    #pragma unroll
    for (int off = 1; off < 32; off <<= 1) {
      float o = __shfl_up(v, off, 32);
      if (lane >= off) v = fminf(v, o);
    }
    float excl = __shfl_up(v, 1, 32);
    if (lane == 0) excl = __builtin_inff();
    float o8[8];
    #pragma unroll
    for (int j = 0; j < 8; ++j) { f[j] = Cf[j] + fminf(g[j], excl); o8[j] = f[j]; }
    float4 o0={o8[0],o8[1],o8[2],o8[3]}, o1={o8[4],o8[5],o8[6],o8[7]};
    *(float4*)(fsb + (t+1)*TS_ + s0) = o0;
    *(float4*)(fsb + (t+1)*TS_ + s0 + 4) = o1;
    sc0 = n0; sc1 = n1;
  }
}

// ---------------------------------------------------------------------------
// Kernel 3: backward DP; writes cs[t] = fs[t] + bs[t] in place over fs buffer.
// bcol is the reference's flipped prefix sum: step t adds -scores[127-t, 255].
// ---------------------------------------------------------------------------
__global__ __launch_bounds__(32) void bs_kernel(
    const float* __restrict__ scores, float* __restrict__ cs)
{
  const int b = blockIdx.x, lane = threadIdx.x;
  const int s0 = lane * 8;
  float Pb[8], bn[8];
  #pragma unroll
  for (int j = 0; j < 8; ++j) {
    float sv = (float)(s0 + j);
    Pb[j] = sv * (sv + 1.f) * 0.5f * (1.f/(float)TS_);
    bn[j] = ((float)TS_ - sv) * (1.f/(float)TS_);       // bs_last
  }
  float* csb = cs + (size_t)b * (TT_+1) * TS_;
  const float* scb = scores + (size_t)b * TT_ * TS_;
  { // cs[TT] = fs[TT] + bs_last
    float4 a0 = *(const float4*)(csb + TT_*TS_ + s0);
    float4 a1 = *(const float4*)(csb + TT_*TS_ + s0 + 4);
    a0.x+=bn[0]; a0.y+=bn[1]; a0.z+=bn[2]; a0.w+=bn[3];
    a1.x+=bn[4]; a1.y+=bn[5]; a1.z+=bn[6]; a1.w+=bn[7];
    *(float4*)(csb + TT_*TS_ + s0) = a0;
    *(float4*)(csb + TT_*TS_ + s0 + 4) = a1;
  }
  float bcsum = 0.f;
  float bq = scb[255];                                   // scores[row 0][255], used at t=127
  float4 sc0 = *(const float4*)(scb + (TT_-1)*TS_ + s0);
  float4 sc1 = *(const float4*)(scb + (TT_-1)*TS_ + s0 + 4);
  float4 ff0 = *(const float4*)(csb + (TT_-1)*TS_ + s0);
  float4 ff1 = *(const float4*)(csb + (TT_-1)*TS_ + s0 + 4);
  for (int t = TT_-1; t >= 0; --t) {
    float4 nsc0={},nsc1={},nff0={},nff1={}; float nbq = 0.f;
    if (t > 0) {                                          // prefetch next iteration
      nsc0 = *(const float4*)(scb + (t-1)*TS_ + s0);
      nsc1 = *(const float4*)(scb + (t-1)*TS_ + s0 + 4);
      nff0 = *(const float4*)(csb + (t-1)*TS_ + s0);
      nff1 = *(const float4*)(csb + (t-1)*TS_ + s0 + 4);
      nbq  = scb[(TT_ - t)*TS_ + 255];                    // row 127-(t-1)
    }
    bcsum += -bq;                                         // sc_dp = -scores
    float scv[8] = {sc0.x,sc0.y,sc0.z,sc0.w,sc1.x,sc1.y,sc1.z,sc1.w};
    float ffv[8] = {ff0.x,ff0.y,ff0.z,ff0.w,ff1.x,ff1.y,ff1.z,ff1.w};
    float hh[8]; float run = __builtin_inff();
    #pragma unroll
    for (int j = 7; j >= 0; --j) {                        // local reverse cummin
      float val = bn[j] - scv[j];
      if (s0 + j == TS_-1) val = bcsum;                   // .at[:,-1].set(bc_t)
      float hv = val + Pb[j];
      run = fminf(run, hv); hh[j] = run;
    }
    float v = run;                                        // wave reverse min-scan
    #pragma unroll
    for (int off = 1; off < 32; off <<= 1) {
      float o = __shfl_down(v, off, 32);
      if (lane < 32 - off) v = fminf(v, o);
    }
    float excl = __shfl_down(v, 1, 32);
    if (lane == 31) excl = __builtin_inff();
    float o8[8];
    #pragma unroll
    for (int j = 0; j < 8; ++j) {
      float brow = fminf(hh[j], excl) - Pb[j];
      bn[j] = brow;
      o8[j] = ffv[j] + brow;                              // cs[t] = fs[t] + bs[t]
    }
    float4 o0={o8[0],o8[1],o8[2],o8[3]}, o1={o8[4],o8[5],o8[6],o8[7]};
    *(float4*)(csb + t*TS_ + s0) = o0;
    *(float4*)(csb + t*TS_ + s0 + 4) = o1;
    sc0=nsc0; sc1=nsc1; ff0=nff0; ff1=nff1; bq=nbq;
  }
}

// ---------------------------------------------------------------------------
// Kernel 4: backtrack, one thread per batch.
// ---------------------------------------------------------------------------
__global__ void backtrack_kernel(const float* __restrict__ cs, int* __restrict__ best)
{
  int b = threadIdx.x;
  if (b >= B_) return;
  const float* csb = cs + (size_t)b * (TT_+1) * TS_;
  int* bb = best + b * TT_;
  for (int t = 0; t < TT_; ++t) bb[t] = TS_ - 1;
  int t = 0, j = 0;
  for (int it = 0; it < TT_ + TS_ - 1; ++it) {
    if (t < TT_ && j < TS_ - 1) {
      if (csb[(t+1)*TS_ + j] <= csb[t*TS_ + j + 1]) { bb[t] = j; ++t; }
      else ++j;
    }
  }
}

// ---------------------------------------------------------------------------
// Kernel 5: gamma / read / write from best_context.
// gamma[b,t,s] = (s >= best[b,t]); write = gamma[:,1:]; read = 1 - write.
// ---------------------------------------------------------------------------
__global__ __launch_bounds__(256) void emit_kernel(
    const int* __restrict__ best, float* __restrict__ out)
{
  int idx = blockIdx.x * 256 + threadIdx.x;
  if (idx >= B_*TT_*TS_) return;
  int s  = idx & (TS_-1);
  int bt = idx >> 8;           // = b*TT_ + t
  int t  = bt & (TT_-1);
  int b  = bt >> 7;
  float g = (s >= best[bt]) ? 1.0f : 0.0f;
  out[GOFF + idx] = g;
  if (t >= 1) {
    int widx = (b*(TT_-1) + (t-1))*TS_ + s;
    out[WOFF + widx] = g;
    out[ROFF + widx] = 1.0f - g;
  }
}

// ---------------------------------------------------------------------------
extern "C" void kernel_launch(void* const* d_in, const int* in_sizes, int n_in,
                              void* d_out, int out_size, void* d_ws, size_t ws_size,
                              hipStream_t stream)
{
  const float* obs    = (const float*)d_in[0];
  const float* scores = (const float*)d_in[1];
  const float* gw     = (const float*)d_in[2];
  const float* gb     = (const float*)d_in[3];
  float* out   = (float*)d_out;
  float* csbuf = (float*)d_ws;                                    // 129*16*256 f32
  int*   best  = (int*)((char*)d_ws + (size_t)B_*(TT_+1)*TS_*4);  // 2048 i32

  // 16*127*16 = 32512 waves, 8 waves per 256-thread block -> 4064 blocks exact
  gate_kernel<<<(B_*(TT_-1)*16)/8, 256, 0, stream>>>(obs, gw, gb, out);
  fs_kernel<<<B_, 32, 0, stream>>>(scores, csbuf);
  bs_kernel<<<B_, 32, 0, stream>>>(scores, csbuf);
  backtrack_kernel<<<1, 32, 0, stream>>>(csbuf, best);
  emit_kernel<<<(B_*TT_*TS_ + 255)/256, 256, 0, stream>>>(best, out);
}